// QuadraticLayer_13125420057031
// MI455X (gfx1250) — compile-verified
//
#include <hip/hip_runtime.h>

typedef __attribute__((ext_vector_type(2))) float v2f;
typedef __attribute__((ext_vector_type(8))) float v8f;

#define B_  4
#define H_  112
#define W_  112
#define C_  48
#define F_  48
#define HO  110
#define WO  110

#define NPAIR_SYM 45                          // unordered (pi,pj) pairs, pi<=pj
#define PAIR_SZ   (F_ * C_)                   // one pair block, layout [f][c] = 2304 floats
#define WSYM_ELEMS (NPAIR_SYM * PAIR_SZ)      // 103680 floats
#define WLT_ELEMS  (9 * PAIR_SZ)              // 20736 floats
#define WS_ELEMS   (WSYM_ELEMS + WLT_ELEMS)   // ~486 KB of workspace

// LDS slab: xs[c][di][wl]; 66 columns cover 64 output pixels + 2 halo.
#define SLAB_W  66
#define SLAB_DI SLAB_W
#define SLAB_C  (3 * SLAB_W)
#define SLAB_ELEMS (C_ * 3 * SLAB_W)          // 9504 floats = 38 KB

// ---------------------------------------------------------------------------
// Repack: Wsym[ps][f][c] = Wv[c,pi,pj,f] (+ Wv[c,pj,pi,f] if pi!=pj)
//         WlT [pos][f][c] = Wl[pos,c,f]
// [f][c] layout makes each B fragment (two adjacent K rows at one f) a single
// 8-byte load.
// ---------------------------------------------------------------------------
__global__ __launch_bounds__(256)
void volterra_repack_kernel(const float* __restrict__ Wv,
                            const float* __restrict__ Wl,
                            float* __restrict__ ws)
{
    const int t = blockIdx.x * 256 + threadIdx.x;
    if (t < WSYM_ELEMS) {
        const int c  = t % C_;
        const int f  = (t / C_) % F_;
        const int ps = t / PAIR_SZ;
        int pi = 0, rem = ps;                  // ps -> (pi,pj), pi-major, pi<=pj
        while (rem >= 9 - pi) { rem -= 9 - pi; ++pi; }
        const int pj = pi + rem;
        float v = Wv[((c * 9 + pi) * 9 + pj) * F_ + f];
        if (pj != pi) v += Wv[((c * 9 + pj) * 9 + pi) * F_ + f];
        ws[t] = v;
    } else if (t < WS_ELEMS) {
        const int u   = t - WSYM_ELEMS;
        const int c   = u % C_;
        const int f   = (u / C_) % F_;
        const int pos = u / PAIR_SZ;
        ws[t] = Wl[(pos * C_ + c) * F_ + f];
    }
}

// One K=4 GEMM step against all three 16-wide F tiles (f0 = 0,16,32).
// B fragment: lane(l) needs rows K=cl,cl+1 at column f0+l%16 -> one b64 load.
__device__ __forceinline__ void mm3(const v2f A, const float* __restrict__ bp,
                                    const int cl, v8f& a0, v8f& a1, v8f& a2)
{
    const v2f B0 = *(const v2f*)(bp + cl);
    const v2f B1 = *(const v2f*)(bp + 16 * C_ + cl);
    const v2f B2 = *(const v2f*)(bp + 32 * C_ + cl);
    a0 = __builtin_amdgcn_wmma_f32_16x16x4_f32(false, A, false, B0, (short)0, a0, false, false);
    a1 = __builtin_amdgcn_wmma_f32_16x16x4_f32(false, A, false, B1, (short)0, a1, false, false);
    a2 = __builtin_amdgcn_wmma_f32_16x16x4_f32(false, A, false, B2, (short)0, a2, false, false);
}

__global__ __launch_bounds__(128)
void volterra_wmma_f32_kernel(const float* __restrict__ x,
                              const float* __restrict__ bias,
                              const float* __restrict__ Wsym, // [45][F][C]
                              const float* __restrict__ WlT,  // [9][F][C]
                              float* __restrict__ out)        // [B][HO][WO][F]
{
    __shared__ float xs[SLAB_ELEMS];

    const int bi = blockIdx.x;
    const int g  = bi % 2;
    const int h  = (bi / 2) % HO;
    const int b  = bi / (2 * HO);
    const int wbase = g * 64;

    // ---- stage x slab into LDS, transposed NHWC -> [c][di][wl] ----
    const int tid = threadIdx.x;
    for (int idx = tid; idx < SLAB_ELEMS; idx += 128) {
        int c    = idx % C_;
        int rest = idx / C_;
        int wl   = rest % SLAB_W;
        int di   = rest / SLAB_W;
        int gw   = wbase + wl;  if (gw > W_ - 1) gw = W_ - 1;   // clamp; masked at store
        xs[c * SLAB_C + di * SLAB_DI + wl] =
            x[((b * H_ + (h + di)) * W_ + gw) * C_ + c];
    }
    __syncthreads();

    const int wv   = tid >> 5;
    const int lane = tid & 31;
    const int m    = lane & 15;
    const int half = lane >> 4;
    const int wloc = wv * 16 + m;
    const int fcol = m * C_;                 // column offset within a [f][c] block

    v8f acc0 = {}; v8f acc1 = {}; v8f acc2 = {};

    int ps = 0;
    for (int pi = 0; pi < 9; ++pi) {
        const int ri = pi / 3, ci = pi % 3;
        const float* rowi = &xs[ri * SLAB_DI + wloc + ci];

        // register-cache this lane's p_i values for all 12 K chunks
        float pv[24];
        #pragma unroll
        for (int t = 0; t < 12; ++t) {
            const int cl = 4 * t + 2 * half;
            pv[2 * t]     = rowi[cl * SLAB_C];
            pv[2 * t + 1] = rowi[(cl + 1) * SLAB_C];
        }

        // ---- linear conv term, pos = pi (A = p_i, no extra LDS traffic) ----
        {
            const float* bp = WlT + pi * PAIR_SZ + fcol;
            #pragma unroll
            for (int t = 0; t < 12; ++t) {
                const int cl = 4 * t + 2 * half;
                v2f A; A.x = pv[2 * t]; A.y = pv[2 * t + 1];
                mm3(A, bp, cl, acc0, acc1, acc2);
            }
        }

        // ---- diagonal quadratic term (pi,pi): A = p_i^2 ----
        {
            const float* bp = Wsym + ps * PAIR_SZ + fcol; ++ps;
            #pragma unroll
            for (int t = 0; t < 12; ++t) {
                const int cl = 4 * t + 2 * half;
                v2f A;
                A.x = pv[2 * t]     * pv[2 * t];
                A.y = pv[2 * t + 1] * pv[2 * t + 1];
                mm3(A, bp, cl, acc0, acc1, acc2);
            }
        }

        // ---- off-diagonal pairs (pi<pj): A = p_i * p_j, B pre-symmetrized ----
        for (int pj = pi + 1; pj < 9; ++pj, ++ps) {
            const int rj = pj / 3, cj = pj % 3;
            const float* rowj = &xs[rj * SLAB_DI + wloc + cj];
            const float* bp = Wsym + ps * PAIR_SZ + fcol;
            __builtin_prefetch(bp + PAIR_SZ, 0, 1);
            #pragma unroll
            for (int t = 0; t < 12; ++t) {
                const int cl = 4 * t + 2 * half;
                v2f A;
                A.x = pv[2 * t]     * rowj[cl * SLAB_C];
                A.y = pv[2 * t + 1] * rowj[(cl + 1) * SLAB_C];
                mm3(A, bp, cl, acc0, acc1, acc2);
            }
        }
    }

    // ---- epilogue: bias + masked store -------------------------------------
    const float bv0 = bias[m];
    const float bv1 = bias[16 + m];
    const float bv2 = bias[32 + m];
    const int wpix_base = wbase + wv * 16 + 8 * half;
    #pragma unroll
    for (int v = 0; v < 8; ++v) {
        const int w = wpix_base + v;
        if (w < WO) {
            float* o = &out[((size_t)(b * HO + h) * WO + w) * F_ + m];
            o[0]  = acc0[v] + bv0;
            o[16] = acc1[v] + bv1;
            o[32] = acc2[v] + bv2;
        }
    }
}

extern "C" void kernel_launch(void* const* d_in, const int* in_sizes, int n_in,
                              void* d_out, int out_size, void* d_ws, size_t ws_size,
                              hipStream_t stream) {
    (void)in_sizes; (void)n_in; (void)out_size; (void)ws_size;
    const float* x  = (const float*)d_in[0];
    const float* Wv = (const float*)d_in[1];
    const float* Wl = (const float*)d_in[2];
    const float* bs = (const float*)d_in[3];
    float* ws  = (float*)d_ws;              // needs WS_ELEMS*4 ~ 498 KB
    float* out = (float*)d_out;

    volterra_repack_kernel<<<(WS_ELEMS + 255) / 256, 256, 0, stream>>>(Wv, Wl, ws);

    dim3 grid(B_ * HO * 2);                 // 880 blocks
    dim3 block(128);                        // 4 waves; each: 16 pixels x 48 features
    volterra_wmma_f32_kernel<<<grid, block, 0, stream>>>(
        x, bs, ws, ws + WSYM_ELEMS, out);
}